// GraphClassificationGCN_27496380629809
// MI455X (gfx1250) — compile-verified
//
#include <hip/hip_runtime.h>
#include <hip/hip_bf16.h>

#define NN 100000
#define EE 1600000
#define FH 128
#define GG 500
#define CC 10
#define NB ((NN + 255) / 256)   // scan blocks = 391

typedef float v2f __attribute__((ext_vector_type(2)));
typedef float v8f __attribute__((ext_vector_type(8)));

// ---- integer in-degree histogram ------------------------------------------
__global__ __launch_bounds__(256) void gcn_deg(const int* __restrict__ dst,
                                               int* __restrict__ degi) {
  int e = blockIdx.x * blockDim.x + threadIdx.x;
  if (e < EE) atomicAdd(&degi[dst[e]], 1);
}

__global__ __launch_bounds__(256) void gcn_dinv(const int* __restrict__ degi,
                                                float* __restrict__ dinv) {
  int i = blockIdx.x * blockDim.x + threadIdx.x;
  if (i < NN) dinv[i] = rsqrtf((float)degi[i] + 1.0f);
}

// ---- two-level exclusive scan of degrees -> CSR row starts -----------------
__global__ __launch_bounds__(256) void scan_local(const int* __restrict__ degi,
                                                  int* __restrict__ rowstart,
                                                  int* __restrict__ blocksums) {
  __shared__ int sm[256];
  int i = blockIdx.x * 256 + threadIdx.x;
  int v = (i < NN) ? degi[i] : 0;
  int acc = v;
  sm[threadIdx.x] = acc;
  __syncthreads();
#pragma unroll
  for (int off = 1; off < 256; off <<= 1) {
    int t = (threadIdx.x >= off) ? sm[threadIdx.x - off] : 0;
    __syncthreads();
    acc += t;
    sm[threadIdx.x] = acc;
    __syncthreads();
  }
  if (i < NN) rowstart[i] = acc - v;                 // exclusive within block
  if (threadIdx.x == 255) blocksums[blockIdx.x] = acc;
}

__global__ __launch_bounds__(512) void scan_blocks(int* __restrict__ blocksums) {
  __shared__ int sm[512];
  int tid = threadIdx.x;
  int v = (tid < NB) ? blocksums[tid] : 0;
  int acc = v;
  sm[tid] = acc;
  __syncthreads();
#pragma unroll
  for (int off = 1; off < 512; off <<= 1) {
    int t = (tid >= off) ? sm[tid - off] : 0;
    __syncthreads();
    acc += t;
    sm[tid] = acc;
    __syncthreads();
  }
  if (tid < NB) blocksums[tid] = acc - v;            // exclusive block offsets
}

__global__ __launch_bounds__(256) void scan_add(int* __restrict__ rowstart,
                                                const int* __restrict__ blockoffs) {
  int i = blockIdx.x * 256 + threadIdx.x;
  if (i < NN) rowstart[i] += blockoffs[blockIdx.x];
}

// ---- CSR fill: csr_src[rowstart[dst] + pos] = src --------------------------
__global__ __launch_bounds__(256) void csr_fill(const int* __restrict__ src,
                                                const int* __restrict__ dst,
                                                const int* __restrict__ rowstart,
                                                int* __restrict__ cursor,
                                                int* __restrict__ csr_src) {
  int e = blockIdx.x * blockDim.x + threadIdx.x;
  if (e < EE) {
    int d = dst[e];
    int pos = atomicAdd(&cursor[d], 1);
    csr_src[rowstart[d] + pos] = src[e];
  }
}

// ---- dense transform: Out = act(A) @ W via V_WMMA_F32_16X16X4_F32 ----------
// One block = one 16-row slab, 8 waves = 8 column tiles. A slab staged in LDS
// (ReLU applied once at stage time); inner loop: ds_load_b64 A-pair + 2x b32 B.
template <int RELU>
__global__ __launch_bounds__(256) void gcn_gemm_wmma(const float* __restrict__ A,
                                                     const float* __restrict__ W,
                                                     float* __restrict__ Out) {
  __shared__ float As[16 * FH];                      // 8 KB
  const int lane  = threadIdx.x & 31;
  const int wave  = threadIdx.x >> 5;
  const int mtile = blockIdx.x;

  {
    const float4* srcv = (const float4*)(A + (long)mtile * 16 * FH);
    float4* dstv = (float4*)As;
#pragma unroll
    for (int i = threadIdx.x; i < 16 * FH / 4; i += 256) {
      float4 v = srcv[i];
      if (RELU) {
        v.x = fmaxf(v.x, 0.0f); v.y = fmaxf(v.y, 0.0f);
        v.z = fmaxf(v.z, 0.0f); v.w = fmaxf(v.w, 0.0f);
      }
      dstv[i] = v;
    }
  }
  __syncthreads();

  const int koff = (lane >> 4) << 1;                 // 0 or 2
  const int col  = (wave << 4) + (lane & 15);        // ntile = wave
  const float* arow = As + (lane & 15) * FH;

  v8f c = {};
#pragma unroll
  for (int k = 0; k < FH; k += 4) {
    v2f a = *(const v2f*)(arow + k + koff);          // ds_load_b64
    v2f b;
    b[0] = W[(k + koff) * FH + col];
    b[1] = W[(k + koff + 1) * FH + col];
    c = __builtin_amdgcn_wmma_f32_16x16x4_f32(false, a, false, b, (short)0, c,
                                              false, false);
  }

  const int orow = (mtile << 4) + ((lane >> 4) << 3);
#pragma unroll
  for (int r = 0; r < 8; ++r)
    Out[(long)(orow + r) * FH + col] = c[r];
}

// ---- atomic-free aggregation: one wave per node, gather in-edges -----------
// acc = h[node]*dinv^2 + bias  (self-loop + bias fused), then sum over CSR.
__global__ __launch_bounds__(256) void gcn_aggregate(const float* __restrict__ h,
                                                     const int* __restrict__ csr_src,
                                                     const int* __restrict__ rowstart,
                                                     const int* __restrict__ degi,
                                                     const float* __restrict__ dinv,
                                                     const float* __restrict__ bias,
                                                     float* __restrict__ outp) {
  long t = (long)blockIdx.x * blockDim.x + threadIdx.x;
  int node = (int)(t >> 5);
  if (node >= NN) return;
  int g = (int)(t & 31);

  float di = dinv[node];
  float sl = di * di;
  float4 hv = ((const float4*)(h + (long)node * FH))[g];
  float4 bv = ((const float4*)bias)[g];
  float4 acc;
  acc.x = hv.x * sl + bv.x;
  acc.y = hv.y * sl + bv.y;
  acc.z = hv.z * sl + bv.z;
  acc.w = hv.w * sl + bv.w;

  int start = rowstart[node];
  int cnt = degi[node];
  for (int j = 0; j < cnt; ++j) {
    int s = csr_src[start + j];
    float nrm = dinv[s] * di;
    float4 v = ((const float4*)(h + (long)s * FH))[g];
    acc.x += v.x * nrm;
    acc.y += v.y * nrm;
    acc.z += v.z * nrm;
    acc.w += v.w * nrm;
  }
  ((float4*)(outp + (long)node * FH))[g] = acc;
}

// ---- pooling: relu + segment mean/max accumulation -------------------------
__global__ __launch_bounds__(256) void gcn_pool(const float* __restrict__ h,
                                                const int* __restrict__ batch,
                                                float* __restrict__ sums,
                                                unsigned int* __restrict__ maxu,
                                                float* __restrict__ counts) {
  long t = (long)blockIdx.x * blockDim.x + threadIdx.x;
  int node = (int)(t >> 5);
  if (node >= NN) return;
  int g = (int)(t & 31);
  int b = batch[node];
  float4 v = ((const float4*)(h + (long)node * FH))[g];
  float r0 = fmaxf(v.x, 0.0f), r1 = fmaxf(v.y, 0.0f);
  float r2 = fmaxf(v.z, 0.0f), r3 = fmaxf(v.w, 0.0f);
  float* so = sums + (long)b * FH + (g << 2);
  atomicAdd(so + 0, r0);
  atomicAdd(so + 1, r1);
  atomicAdd(so + 2, r2);
  atomicAdd(so + 3, r3);
  // post-ReLU values are >= 0, so bit-pattern max on unsigned is order-correct
  unsigned int* mo = maxu + (long)b * FH + (g << 2);
  atomicMax(mo + 0, __float_as_uint(r0));
  atomicMax(mo + 1, __float_as_uint(r1));
  atomicMax(mo + 2, __float_as_uint(r2));
  atomicMax(mo + 3, __float_as_uint(r3));
  if (g == 0) atomicAdd(&counts[b], 1.0f);
}

// ---- final FC: [G, 2H] @ [2H, C] + b ---------------------------------------
__global__ __launch_bounds__(256) void gcn_fc(const float* __restrict__ sums,
                                              const unsigned int* __restrict__ maxu,
                                              const float* __restrict__ counts,
                                              const float* __restrict__ fcW,
                                              const float* __restrict__ fcb,
                                              float* __restrict__ out) {
  int idx = blockIdx.x * blockDim.x + threadIdx.x;
  if (idx >= GG * CC) return;
  int g = idx / CC, c = idx % CC;
  float inv = 1.0f / fmaxf(counts[g], 1.0f);
  float acc = fcb[c];
  const float* srow = sums + (long)g * FH;
  const unsigned int* mrow = maxu + (long)g * FH;
#pragma unroll 8
  for (int f = 0; f < FH; ++f) acc += (srow[f] * inv) * fcW[f * CC + c];
#pragma unroll 8
  for (int f = 0; f < FH; ++f) acc += __uint_as_float(mrow[f]) * fcW[(FH + f) * CC + c];
  out[idx] = acc;
}

extern "C" void kernel_launch(void* const* d_in, const int* in_sizes, int n_in,
                              void* d_out, int out_size, void* d_ws, size_t ws_size,
                              hipStream_t stream) {
  (void)in_sizes; (void)n_in; (void)out_size; (void)ws_size;

  const float* x   = (const float*)d_in[0];
  const int* ei    = (const int*)d_in[1];     // [2, E] flat: src then dst
  const int* src   = ei;
  const int* dstp  = ei + EE;
  const int* batch = (const int*)d_in[2];
  const float* W1  = (const float*)d_in[3];
  const float* b1  = (const float*)d_in[4];
  const float* W2  = (const float*)d_in[5];
  const float* b2  = (const float*)d_in[6];
  const float* W3  = (const float*)d_in[7];
  const float* b3  = (const float*)d_in[8];
  const float* fcW = (const float*)d_in[9];
  const float* fcb = (const float*)d_in[10];
  float* out = (float*)d_out;

  // workspace layout:
  // [degi N][cursor N][rowstart N][blocksums NB][csr_src E] (ints)
  // [dinv N][bufA N*H][bufB N*H][sums G*H][maxu G*H][counts G] (floats)
  int* degi      = (int*)d_ws;
  int* cursor    = degi + NN;
  int* rowstart  = cursor + NN;
  int* blocksums = rowstart + NN;
  int* csr_src   = blocksums + NB;
  float* dinv    = (float*)(csr_src + EE);
  float* bufA    = dinv + NN;
  float* bufB    = bufA + (size_t)NN * FH;
  float* sums    = bufB + (size_t)NN * FH;
  unsigned int* maxu = (unsigned int*)(sums + (size_t)GG * FH);
  float* counts  = (float*)(maxu + (size_t)GG * FH);

  hipMemsetAsync(degi, 0, (size_t)2 * NN * sizeof(int), stream);   // degi + cursor
  hipMemsetAsync(sums, 0, (size_t)(2 * GG * FH + GG) * sizeof(float), stream);

  // ---- build degree, norm, CSR (reused by all 3 layers) ----
  gcn_deg<<<(EE + 255) / 256, 256, 0, stream>>>(dstp, degi);
  gcn_dinv<<<(NN + 255) / 256, 256, 0, stream>>>(degi, dinv);
  scan_local<<<NB, 256, 0, stream>>>(degi, rowstart, blocksums);
  scan_blocks<<<1, 512, 0, stream>>>(blocksums);
  scan_add<<<NB, 256, 0, stream>>>(rowstart, blocksums);
  csr_fill<<<(EE + 255) / 256, 256, 0, stream>>>(src, dstp, rowstart, cursor, csr_src);

  const int gemm_blocks = NN / 16;                        // 6250
  const int agg_blocks  = (int)(((long)NN * 32 + 255) / 256);

  // ---- Layer 1 ----
  gcn_gemm_wmma<0><<<gemm_blocks, 256, 0, stream>>>(x, W1, bufA);
  gcn_aggregate<<<agg_blocks, 256, 0, stream>>>(bufA, csr_src, rowstart, degi,
                                                dinv, b1, bufB);
  // ---- Layer 2 (ReLU fused into GEMM's LDS staging) ----
  gcn_gemm_wmma<1><<<gemm_blocks, 256, 0, stream>>>(bufB, W2, bufA);
  gcn_aggregate<<<agg_blocks, 256, 0, stream>>>(bufA, csr_src, rowstart, degi,
                                                dinv, b2, bufB);
  // ---- Layer 3 ----
  gcn_gemm_wmma<1><<<gemm_blocks, 256, 0, stream>>>(bufB, W3, bufA);
  gcn_aggregate<<<agg_blocks, 256, 0, stream>>>(bufA, csr_src, rowstart, degi,
                                                dinv, b3, bufB);

  // ---- Pool + classify (ReLU fused into pool read) ----
  gcn_pool<<<agg_blocks, 256, 0, stream>>>(bufB, batch, sums, maxu, counts);
  gcn_fc<<<(GG * CC + 255) / 256, 256, 0, stream>>>(sums, maxu, counts, fcW, fcb, out);
}